// MeshGraphEdgeMLPSum_21114059227464
// MI455X (gfx1250) — compile-verified
//
#include <hip/hip_runtime.h>

typedef float v2f __attribute__((ext_vector_type(2)));
typedef float v8f __attribute__((ext_vector_type(8)));

#define LDSTRIDE 132      // 128 + 4 floats pad -> conflict-free b64 A-frag reads
#define EDGE_WAVES 2      // waves per block in edge kernel (2 LDS bufs per wave)
#define NODE_WAVES 4      // waves per block in node kernel (no LDS)

static __device__ __forceinline__ float silu_f(float x) {
    return x * (1.0f / (1.0f + __expf(-x)));
}

// ---------------------------------------------------------------------------
// Kernel 1: node projections  node_src = nfeat @ w_src^T
//                             node_dst = nfeat @ w_dst^T + b1
// One wave = 16 nodes. Four passes of 4 WMMA accumulator tiles each.
// ---------------------------------------------------------------------------
__global__ void node_proj_kernel(const float* __restrict__ nfeat,
                                 const float* __restrict__ w_src,
                                 const float* __restrict__ w_dst,
                                 const float* __restrict__ b1,
                                 float* __restrict__ node_src,
                                 float* __restrict__ node_dst,
                                 int N)
{
    const int lane = threadIdx.x & 31;
    const int wave = threadIdx.x >> 5;
    const int nbase = (blockIdx.x * NODE_WAVES + wave) * 16;
    if (nbase >= N) return;                 // wave-uniform exit

    const int lh   = lane & 15;
    const int half = lane >> 4;
    const int kh   = half * 2;
    const bool fullTile = (nbase + 16 <= N);   // wave-uniform

    int arow = nbase + lh; if (arow >= N) arow = N - 1;
    const int aoff = arow * 128 + kh;       // 32-bit offset

    const v8f zero8 = {0.f,0.f,0.f,0.f,0.f,0.f,0.f,0.f};

    // ---- passes over w_src ----
#pragma unroll 1
    for (int cb = 0; cb < 8; cb += 4) {
        v8f acc[4];
#pragma unroll
        for (int j = 0; j < 4; ++j) acc[j] = zero8;

#pragma unroll 4
        for (int k = 0; k < 128; k += 4) {
            v2f a = *(const v2f*)(nfeat + aoff + k);
#pragma unroll
            for (int j = 0; j < 4; ++j) {
                v2f b = *(const v2f*)(w_src + ((cb + j) * 16 + lh) * 128 + kh + k);
                acc[j] = __builtin_amdgcn_wmma_f32_16x16x4_f32(false, a, false, b,
                                                               (short)0, acc[j], false, false);
            }
        }
        if (fullTile) {
#pragma unroll
            for (int v = 0; v < 8; ++v) {
                const int r = nbase + v + 8 * half;
#pragma unroll
                for (int j = 0; j < 4; ++j)
                    node_src[(size_t)r * 128 + (cb + j) * 16 + lh] = acc[j][v];
            }
        } else {
#pragma unroll
            for (int v = 0; v < 8; ++v) {
                const int r = nbase + v + 8 * half;
                if (r < N) {
#pragma unroll
                    for (int j = 0; j < 4; ++j)
                        node_src[(size_t)r * 128 + (cb + j) * 16 + lh] = acc[j][v];
                }
            }
        }
    }

    // ---- passes over w_dst (+ b1) ----
#pragma unroll 1
    for (int cb = 0; cb < 8; cb += 4) {
        v8f acc[4];
#pragma unroll
        for (int j = 0; j < 4; ++j) acc[j] = zero8;

#pragma unroll 4
        for (int k = 0; k < 128; k += 4) {
            v2f a = *(const v2f*)(nfeat + aoff + k);
#pragma unroll
            for (int j = 0; j < 4; ++j) {
                v2f b = *(const v2f*)(w_dst + ((cb + j) * 16 + lh) * 128 + kh + k);
                acc[j] = __builtin_amdgcn_wmma_f32_16x16x4_f32(false, a, false, b,
                                                               (short)0, acc[j], false, false);
            }
        }
        float b1v[4];
#pragma unroll
        for (int j = 0; j < 4; ++j) b1v[j] = b1[(cb + j) * 16 + lh];

        if (fullTile) {
#pragma unroll
            for (int v = 0; v < 8; ++v) {
                const int r = nbase + v + 8 * half;
#pragma unroll
                for (int j = 0; j < 4; ++j)
                    node_dst[(size_t)r * 128 + (cb + j) * 16 + lh] = acc[j][v] + b1v[j];
            }
        } else {
#pragma unroll
            for (int v = 0; v < 8; ++v) {
                const int r = nbase + v + 8 * half;
                if (r < N) {
#pragma unroll
                    for (int j = 0; j < 4; ++j)
                        node_dst[(size_t)r * 128 + (cb + j) * 16 + lh] = acc[j][v] + b1v[j];
                }
            }
        }
    }
}

// ---------------------------------------------------------------------------
// Kernel 2: per-edge fused  GEMM1 -> +gather(src)+gather(dst) -> SiLU ->
//                           GEMM2 -> +b2 -> LayerNorm -> out
// One wave owns 16 edges; two wave-private padded LDS buffers.
// Gather rows are prefetched before GEMM1 so WMMA work hides their latency.
// ---------------------------------------------------------------------------
__global__ void edge_fused_kernel(const float* __restrict__ efeat,
                                  const int*   __restrict__ src_idx,
                                  const int*   __restrict__ dst_idx,
                                  const float* __restrict__ w_efeat,
                                  const float* __restrict__ w2,
                                  const float* __restrict__ b2,
                                  const float* __restrict__ ln_gamma,
                                  const float* __restrict__ ln_beta,
                                  const float* __restrict__ node_src,
                                  const float* __restrict__ node_dst,
                                  float* __restrict__ out,
                                  int E)
{
    __shared__ float lds[EDGE_WAVES * 2 * 16 * LDSTRIDE];   // 33,792 B

    const int lane = threadIdx.x & 31;
    const int wave = threadIdx.x >> 5;
    const int ebase = (blockIdx.x * EDGE_WAVES + wave) * 16;
    if (ebase >= E) return;                 // wave-uniform exit

    float* bufA = lds + wave * (2 * 16 * LDSTRIDE);
    float* bufB = bufA + 16 * LDSTRIDE;

    const int lh   = lane & 15;
    const int half = lane >> 4;
    const int kh   = half * 2;

    const v8f zero8 = {0.f,0.f,0.f,0.f,0.f,0.f,0.f,0.f};

    // ---- load gather indices early (one per lane-of-16) ----
    int eclampL = ebase + lh; if (eclampL >= E) eclampL = E - 1;
    const int si = src_idx[eclampL];
    const int di = dst_idx[eclampL];

    // ---- stage efeat tile -> bufA (coalesced b128 per row) ----
#pragma unroll 1
    for (int e = 0; e < 16; ++e) {
        int r = ebase + e; if (r >= E) r = E - 1;
        float4 v = *(const float4*)(efeat + (size_t)r * 128 + lane * 4);
        *(float4*)(bufA + e * LDSTRIDE + lane * 4) = v;
    }

    // ---- prefetch gathered node rows (4 x 128B cachelines per 512B row);
    //      lane l covers edge (l&15), cachelines {2*(l>>4), 2*(l>>4)+1} ----
    {
        const size_t so = (size_t)si * 128 + half * 64;
        const size_t do_ = (size_t)di * 128 + half * 64;
        __builtin_prefetch(node_src + so, 0, 3);
        __builtin_prefetch(node_src + so + 32, 0, 3);
        __builtin_prefetch(node_dst + do_, 0, 3);
        __builtin_prefetch(node_dst + do_ + 32, 0, 3);
    }

    // ---- GEMM1: bufA @ w_efeat^T -> bufB ----
    const float* aA = bufA + lh * LDSTRIDE + kh;
#pragma unroll 1
    for (int cb = 0; cb < 8; cb += 4) {
        v8f acc[4];
#pragma unroll
        for (int j = 0; j < 4; ++j) acc[j] = zero8;

#pragma unroll 4
        for (int k = 0; k < 128; k += 4) {
            v2f a = *(const v2f*)(aA + k);
#pragma unroll
            for (int j = 0; j < 4; ++j) {
                v2f b = *(const v2f*)(w_efeat + ((cb + j) * 16 + lh) * 128 + kh + k);
                acc[j] = __builtin_amdgcn_wmma_f32_16x16x4_f32(false, a, false, b,
                                                               (short)0, acc[j], false, false);
            }
        }
#pragma unroll
        for (int j = 0; j < 4; ++j)
#pragma unroll
            for (int v = 0; v < 8; ++v)
                bufB[(v + 8 * half) * LDSTRIDE + (cb + j) * 16 + lh] = acc[j][v];
    }

    // ---- gather node projections, add, SiLU (row-wise, coalesced b128) ----
#pragma unroll 1
    for (int e = 0; e < 16; ++e) {
        const int s = __shfl(si, e, 32);
        const int d = __shfl(di, e, 32);
        const float4 g = *(const float4*)(node_src + (size_t)s * 128 + lane * 4);
        const float4 h = *(const float4*)(node_dst + (size_t)d * 128 + lane * 4);
        float4 x = *(float4*)(bufB + e * LDSTRIDE + lane * 4);
        x.x = silu_f(x.x + g.x + h.x);
        x.y = silu_f(x.y + g.y + h.y);
        x.z = silu_f(x.z + g.z + h.z);
        x.w = silu_f(x.w + g.w + h.w);
        *(float4*)(bufB + e * LDSTRIDE + lane * 4) = x;
    }

    // ---- GEMM2: bufB @ w2^T (+ b2) -> bufA ----
    const float* aB = bufB + lh * LDSTRIDE + kh;
#pragma unroll 1
    for (int cb = 0; cb < 8; cb += 4) {
        v8f acc[4];
#pragma unroll
        for (int j = 0; j < 4; ++j) acc[j] = zero8;

#pragma unroll 4
        for (int k = 0; k < 128; k += 4) {
            v2f a = *(const v2f*)(aB + k);
#pragma unroll
            for (int j = 0; j < 4; ++j) {
                v2f b = *(const v2f*)(w2 + ((cb + j) * 16 + lh) * 128 + kh + k);
                acc[j] = __builtin_amdgcn_wmma_f32_16x16x4_f32(false, a, false, b,
                                                               (short)0, acc[j], false, false);
            }
        }
        float b2v[4];
#pragma unroll
        for (int j = 0; j < 4; ++j) b2v[j] = b2[(cb + j) * 16 + lh];
#pragma unroll
        for (int j = 0; j < 4; ++j)
#pragma unroll
            for (int v = 0; v < 8; ++v)
                bufA[(v + 8 * half) * LDSTRIDE + (cb + j) * 16 + lh] = acc[j][v] + b2v[j];
    }

    // ---- LayerNorm on bufA rows + coalesced b128 store ----
    const float4 gv = *(const float4*)(ln_gamma + lane * 4);
    const float4 bv = *(const float4*)(ln_beta + lane * 4);
    const bool fullTile = (ebase + 16 <= E);
#pragma unroll 1
    for (int e = 0; e < 16; ++e) {
        float4 x = *(float4*)(bufA + e * LDSTRIDE + lane * 4);
        float s = x.x + x.y + x.z + x.w;
        float q = x.x * x.x + x.y * x.y + x.z * x.z + x.w * x.w;
#pragma unroll
        for (int m = 1; m <= 16; m <<= 1) {
            s += __shfl_xor(s, m, 32);
            q += __shfl_xor(q, m, 32);
        }
        const float mean = s * (1.0f / 128.0f);
        const float var  = q * (1.0f / 128.0f) - mean * mean;
        const float rstd = rsqrtf(var + 1e-5f);

        float4 o;
        o.x = (x.x - mean) * rstd * gv.x + bv.x;
        o.y = (x.y - mean) * rstd * gv.y + bv.y;
        o.z = (x.z - mean) * rstd * gv.z + bv.z;
        o.w = (x.w - mean) * rstd * gv.w + bv.w;

        const int r = ebase + e;
        if (fullTile || r < E)
            *(float4*)(out + (size_t)r * 128 + lane * 4) = o;
    }
}

// ---------------------------------------------------------------------------
extern "C" void kernel_launch(void* const* d_in, const int* in_sizes, int n_in,
                              void* d_out, int out_size, void* d_ws, size_t ws_size,
                              hipStream_t stream)
{
    const float* efeat    = (const float*)d_in[0];
    const float* nfeat    = (const float*)d_in[1];
    const int*   src_idx  = (const int*)  d_in[2];
    const int*   dst_idx  = (const int*)  d_in[3];
    const float* w_efeat  = (const float*)d_in[4];
    const float* w_src    = (const float*)d_in[5];
    const float* w_dst    = (const float*)d_in[6];
    const float* b1       = (const float*)d_in[7];
    const float* w2       = (const float*)d_in[8];
    const float* b2       = (const float*)d_in[9];
    const float* ln_gamma = (const float*)d_in[10];
    const float* ln_beta  = (const float*)d_in[11];
    float* out = (float*)d_out;

    const int E = in_sizes[0] / 128;
    const int N = in_sizes[1] / 128;

    float* node_src = (float*)d_ws;                    // [N,128]
    float* node_dst = node_src + (size_t)N * 128;      // [N,128]

    const int nodeRows = NODE_WAVES * 16;              // 64
    const int edgeRows = EDGE_WAVES * 16;              // 32

    node_proj_kernel<<<(N + nodeRows - 1) / nodeRows, NODE_WAVES * 32, 0, stream>>>(
        nfeat, w_src, w_dst, b1, node_src, node_dst, N);

    edge_fused_kernel<<<(E + edgeRows - 1) / edgeRows, EDGE_WAVES * 32, 0, stream>>>(
        efeat, src_idx, dst_idx, w_efeat, w2, b2, ln_gamma, ln_beta,
        node_src, node_dst, out, E);
}